// GraphSortPooling_82729660056048
// MI455X (gfx1250) — compile-verified
//
#include <hip/hip_runtime.h>
#include <hip/hip_bf16.h>
#include <math.h>
#include <stdint.h>

// Problem constants (match reference)
constexpr int kB = 256;
constexpr int kN = 1024;
constexpr int kH = 256;
constexpr int kK = 5;
constexpr float kNegSlope = 0.01f;

// ---------------------------------------------------------------------------
// Pass 1: node_max[b*N+n] = max over H of h[b,n,:]
// Double-buffered CDNA5 async pipeline: stream 32 KB tiles of h into LDS with
// global_load_async_to_lds_b128 (ASYNCcnt, in-order completion), overlapping
// the DMA of tile t+1 with the LDS max-reduction of tile t.
// Total traffic: 268 MB -> ~11.5 us at 23.3 TB/s; everything else is noise.
// ---------------------------------------------------------------------------
constexpr int kTileRows     = 32;                 // 32 rows x 1 KB = 32 KB tile
constexpr int kTileFloats   = kTileRows * kH;     // 8192 floats
constexpr int kTilesPerBlk  = 8;                  // 8 tiles per block
constexpr int kP1Blocks     = (kB * kN) / (kTileRows * kTilesPerBlk);  // 1024

__global__ void __launch_bounds__(256)
pass1_node_max(const float* __restrict__ h, float* __restrict__ node_max) {
    __shared__ __align__(16) float smem[2][kTileFloats];   // 2 x 32 KB
    const int t = threadIdx.x;
    const uint32_t lds0 = (uint32_t)(uintptr_t)(&smem[0][0]);
    const uint32_t lds1 = (uint32_t)(uintptr_t)(&smem[1][0]);
    const size_t blockBase = (size_t)blockIdx.x * (kTilesPerBlk * kTileFloats);

    // Issue one 32 KB tile as 8 async b128 copies per thread (2048 chunks).
    auto issue_tile = [&](int buf, int tile) {
        const float* gb = h + blockBase + (size_t)tile * kTileFloats;
        const uint32_t lb = buf ? lds1 : lds0;
#pragma unroll
        for (int i = 0; i < 8; ++i) {
            const int chunk = i * 256 + t;                 // 16B chunk id, 0..2047
            const float* g = gb + (size_t)chunk * 4;
            const uint32_t l = lb + (uint32_t)chunk * 16u;
            asm volatile("global_load_async_to_lds_b128 %0, %1, off"
                         :: "v"(l), "v"(g) : "memory");
        }
    };

    issue_tile(0, 0);                                      // prologue

    const int wave = t >> 5;
    const int lane = t & 31;

    for (int tile = 0; tile < kTilesPerBlk; ++tile) {
        if (tile + 1 < kTilesPerBlk) {
            issue_tile((tile + 1) & 1, tile + 1);
            // 8 new copies in flight; async completion is in-order, so
            // ASYNCcnt <= 8 guarantees tile `tile` has fully landed in LDS
            // while tile+1 continues streaming.
            asm volatile("s_wait_asynccnt 0x8" ::: "memory");
        } else {
            asm volatile("s_wait_asynccnt 0x0" ::: "memory");
        }
        __syncthreads();                                   // all waves' copies visible

        const float* buf = smem[tile & 1];
#pragma unroll
        for (int r = 0; r < kTileRows / 8; ++r) {          // 8 waves x 4 rows
            const int row = wave * (kTileRows / 8) + r;
            float m = -__builtin_inff();
#pragma unroll
            for (int j = 0; j < kH / 32; ++j)              // conflict-free stride-32
                m = fmaxf(m, buf[row * kH + lane + j * 32]);
#pragma unroll
            for (int off = 16; off > 0; off >>= 1)
                m = fmaxf(m, __shfl_xor(m, off, 32));
            if (lane == 0)
                node_max[(size_t)blockIdx.x * (kTilesPerBlk * kTileRows)
                         + tile * kTileRows + row] = m;
        }
        __syncthreads();   // buffer free before it is re-issued next iteration
    }
}

// ---------------------------------------------------------------------------
// Pass 2: per batch, masked top-K with lowest-index tie-break (== jax top_k).
// One 256-thread block per batch scans 1024 node maxes, K argmax rounds.
// ---------------------------------------------------------------------------
__global__ void __launch_bounds__(256)
pass2_topk(const float* __restrict__ node_max, const int* __restrict__ n_nodes,
           int* __restrict__ topidx, float* __restrict__ topval) {
    __shared__ float sval[256];
    __shared__ int   sidx[256];
    __shared__ int   swin;
    const int b = blockIdx.x;
    const int t = threadIdx.x;
    const int nn = n_nodes[b];

    int sel[kK];
#pragma unroll
    for (int k = 0; k < kK; ++k) sel[k] = -1;

    for (int k = 0; k < kK; ++k) {
        float bestv = -__builtin_inff();
        int   besti = 0x7fffffff;
#pragma unroll
        for (int c = 0; c < kN / 256; ++c) {
            const int j = t + c * 256;
            bool excl = false;
#pragma unroll
            for (int p = 0; p < kK; ++p) excl = excl || (p < k && sel[p] == j);
            const float v = (j < nn) ? node_max[(size_t)b * kN + j]
                                     : -__builtin_inff();
            if (!excl && (v > bestv || (v == bestv && j < besti))) {
                bestv = v; besti = j;
            }
        }
        sval[t] = bestv; sidx[t] = besti;
        __syncthreads();
        for (int s = 128; s > 0; s >>= 1) {
            if (t < s) {
                const float ov = sval[t + s]; const int oi = sidx[t + s];
                if (ov > sval[t] || (ov == sval[t] && oi < sidx[t])) {
                    sval[t] = ov; sidx[t] = oi;
                }
            }
            __syncthreads();
        }
        if (t == 0) {
            swin = sidx[0];
            topidx[b * kK + k] = sidx[0];
            topval[b * kK + k] = sval[0];
        }
        __syncthreads();
        sel[k] = swin;
        __syncthreads();
    }
}

// ---------------------------------------------------------------------------
// Pass 3: per batch — bitonic-sort (ascending) the K selected rows, zero the
// invalid (-inf) ones, compute leaky-relu logits, softmax over K, weighted sum.
// Block = 256 threads = one channel each.
// ---------------------------------------------------------------------------
__device__ __forceinline__ float block_sum(float v, float* sbuf, int t) {
    sbuf[t] = v;
    __syncthreads();
    for (int s = 128; s > 0; s >>= 1) {
        if (t < s) sbuf[t] += sbuf[t + s];
        __syncthreads();
    }
    const float r = sbuf[0];
    __syncthreads();
    return r;
}

__global__ void __launch_bounds__(256)
pass3_attend(const float* __restrict__ h, const float* __restrict__ q,
             const float* __restrict__ W, const int* __restrict__ topidx,
             const float* __restrict__ topval, float* __restrict__ out) {
    __shared__ float s_sort[kH];
    __shared__ float s_red[256];
    const int b = blockIdx.x;
    const int t = threadIdx.x;

    const float w1 = W[t];         // weight on pooled features
    const float w2 = W[kH + t];    // weight on query half
    const float qv = q[(size_t)b * kH + t];
    const float qdot = block_sum(w2 * qv, s_red, t);   // shared across all k

    float outk[kK];
    float logit[kK];

    for (int k = 0; k < kK; ++k) {
        const int idx = topidx[b * kK + k];
        const float tv = topval[b * kK + k];
        const bool valid = isfinite(tv);               // -inf => padded slot
        float v = valid ? h[((size_t)b * kN + idx) * kH + t] : 0.0f;

        // Bitonic sort ascending, 256 elements, one per thread, in LDS.
        s_sort[t] = v;
        __syncthreads();
        for (int kk = 2; kk <= 256; kk <<= 1) {
            for (int j = kk >> 1; j > 0; j >>= 1) {
                const int ixj = t ^ j;
                if (ixj > t) {
                    const float a = s_sort[t];
                    const float c = s_sort[ixj];
                    const bool up = ((t & kk) == 0);
                    if (up ? (a > c) : (a < c)) {
                        s_sort[t] = c; s_sort[ixj] = a;
                    }
                }
                __syncthreads();
            }
        }
        const float sv = s_sort[t];
        __syncthreads();
        outk[k] = sv;

        const float d = block_sum(w1 * sv, s_red, t);
        const float l = d + qdot;
        logit[k] = (l >= 0.0f) ? l : kNegSlope * l;    // leaky_relu
    }

    // Softmax over K (all threads hold identical logits via broadcast reduce).
    float m = logit[0];
#pragma unroll
    for (int k = 1; k < kK; ++k) m = fmaxf(m, logit[k]);
    float e[kK];
    float se = 0.0f;
#pragma unroll
    for (int k = 0; k < kK; ++k) { e[k] = expf(logit[k] - m); se += e[k]; }
    const float inv = 1.0f / se;
    float acc = 0.0f;
#pragma unroll
    for (int k = 0; k < kK; ++k) acc += outk[k] * (e[k] * inv);

    out[(size_t)b * kH + t] = acc;
}

// ---------------------------------------------------------------------------
extern "C" void kernel_launch(void* const* d_in, const int* in_sizes, int n_in,
                              void* d_out, int out_size, void* d_ws, size_t ws_size,
                              hipStream_t stream) {
    const float* h       = (const float*)d_in[0];   // [B,N,H] f32
    const int*   n_nodes = (const int*)d_in[1];     // [B] i32
    const float* q       = (const float*)d_in[2];   // [B,H] f32
    const float* W       = (const float*)d_in[3];   // [1,2H] f32
    float* out = (float*)d_out;                     // [B,H] f32

    // Workspace layout: node_max [B*N] f32 | topidx [B*K] i32 | topval [B*K] f32
    float* node_max = (float*)d_ws;
    int*   topidx   = (int*)((char*)d_ws + (size_t)kB * kN * sizeof(float));
    float* topval   = (float*)((char*)d_ws + (size_t)kB * kN * sizeof(float)
                                           + (size_t)kB * kK * sizeof(int));

    pass1_node_max<<<kP1Blocks, 256, 0, stream>>>(h, node_max);
    pass2_topk<<<kB, 256, 0, stream>>>(node_max, n_nodes, topidx, topval);
    pass3_attend<<<kB, 256, 0, stream>>>(h, q, W, topidx, topval, out);
}